// GATLSTM_41927470743994
// MI455X (gfx1250) — compile-verified
//
#include <hip/hip_runtime.h>
#include <hip/hip_bf16.h>
#include <cstddef>

// ---------------------------------------------------------------------------
// GAT(2 layers) -> LSTM(2 layers) -> 4 FC heads for gfx1250 (MI455X).
// Dominant work: X(96x36480) @ Wih0^T(36480x1000) = 7 GFLOP, Wih0 = 146MB
// (read once, fits in 192MB L2). Done with native fp32 WMMA
// v_wmma_f32_16x16x4_f32 (exact fp32 on the matrix pipe).
// GEMM inner loop: branch-free, b64 fragment loads, 4-way N register
// blocking (1 A frag -> 4 WMMA), K remainder handled in a single guarded
// epilogue iteration.
// ---------------------------------------------------------------------------

#define T_ 96
#define N_ 1140
#define F_ 64
#define H_ 32
#define E_ 22800
#define ET_ (E_ + N_)          // + self loops
#define LH_ 250
#define G4_ 1000               // 4*LH
#define LDG_ 1008              // padded gate width (63 * 16)
#define OUT_ 2280

typedef float v2f __attribute__((ext_vector_type(2)));
typedef float v8f __attribute__((ext_vector_type(8)));

// ---- monotone float<->uint encoding for atomicMax on floats ---------------
__device__ __forceinline__ unsigned encF(float f) {
  unsigned u = __float_as_uint(f);
  return (u & 0x80000000u) ? ~u : (u | 0x80000000u);
}
__device__ __forceinline__ float decF(unsigned u) {
  u = (u & 0x80000000u) ? (u & 0x7FFFFFFFu) : ~u;
  return __uint_as_float(u);
}
#define ENC_NEG_INF 0x007FFFFFu   // encF(-inf)

// ---------------------------------------------------------------------------
// fills
// ---------------------------------------------------------------------------
__global__ void fill_f32(float* p, float v, size_t n) {
  size_t i = (size_t)blockIdx.x * blockDim.x + threadIdx.x;
  if (i < n) p[i] = v;
}
__global__ void fill_u32(unsigned* p, unsigned v, size_t n) {
  size_t i = (size_t)blockIdx.x * blockDim.x + threadIdx.x;
  if (i < n) p[i] = v;
}

// ---------------------------------------------------------------------------
// GAT node projection: HF = X @ W (Fin x 32), plus ssrc = HF.a_s, sdst = HF.a_d
// block (32, 8): one wave per node (H_ == warpSize == 32 on gfx1250).
// ---------------------------------------------------------------------------
__global__ void gat_proj(const float* __restrict__ X, int Fin,
                         const float* __restrict__ W,
                         const float* __restrict__ a_s,
                         const float* __restrict__ a_d,
                         float* __restrict__ HF,
                         float* __restrict__ ssrc,
                         float* __restrict__ sdst) {
  int j = threadIdx.x;                               // feature 0..31
  int n = blockIdx.x * blockDim.y + threadIdx.y;
  int t = blockIdx.y;
  if (n >= N_) return;
  size_t gn = (size_t)t * N_ + n;
  const float* xr = X + gn * (size_t)Fin;
  float acc = 0.f;
  for (int k = 0; k < Fin; ++k) acc += xr[k] * W[k * H_ + j];
  HF[gn * H_ + j] = acc;
  float vs = acc * a_s[j];
  float vd = acc * a_d[j];
  #pragma unroll
  for (int off = 16; off > 0; off >>= 1) {
    vs += __shfl_xor(vs, off, 32);
    vd += __shfl_xor(vd, off, 32);
  }
  if (j == 0) { ssrc[gn] = vs; sdst[gn] = vd; }
}

__device__ __forceinline__ void edge_sd(const int* EI, int t, int e, int& s, int& d) {
  if (e < E_) {
    s = EI[(size_t)t * 2 * E_ + e];
    d = EI[(size_t)t * 2 * E_ + E_ + e];
  } else {
    s = d = e - E_;   // self loop
  }
}

// per-edge score + segment max (encoded atomicMax)
__global__ void edge_score_max(const int* __restrict__ EI,
                               const float* __restrict__ ssrc,
                               const float* __restrict__ sdst,
                               float* __restrict__ eBuf,
                               unsigned* __restrict__ mEnc) {
  int e = blockIdx.x * blockDim.x + threadIdx.x;
  int t = blockIdx.y;
  if (e >= ET_) return;
  int s, d; edge_sd(EI, t, e, s, d);
  float sc = ssrc[(size_t)t * N_ + s] + sdst[(size_t)t * N_ + d];
  float lr = sc > 0.f ? sc : 0.2f * sc;              // leaky_relu(0.2)
  eBuf[(size_t)t * ET_ + e] = lr;
  atomicMax(&mEnc[(size_t)t * N_ + d], encF(lr));
}

// ex = exp(e - m[dst]); segment sum
__global__ void edge_exp(const int* __restrict__ EI,
                         const float* __restrict__ eBuf,
                         const unsigned* __restrict__ mEnc,
                         float* __restrict__ denom,
                         float* __restrict__ exBuf) {
  int e = blockIdx.x * blockDim.x + threadIdx.x;
  int t = blockIdx.y;
  if (e >= ET_) return;
  int s, d; edge_sd(EI, t, e, s, d);
  float m = decF(mEnc[(size_t)t * N_ + d]);
  float ex = expf(eBuf[(size_t)t * ET_ + e] - m);
  exBuf[(size_t)t * ET_ + e] = ex;
  atomicAdd(&denom[(size_t)t * N_ + d], ex);
}

// out[dst] += alpha * HF[src]; one wave per edge (lane = feature)
__global__ void edge_agg(const int* __restrict__ EI,
                         const float* __restrict__ exBuf,
                         const float* __restrict__ denom,
                         const float* __restrict__ HF,
                         float* __restrict__ outBuf) {
  int j = threadIdx.x;
  int e = blockIdx.x * blockDim.y + threadIdx.y;
  int t = blockIdx.y;
  if (e >= ET_) return;
  int s, d; edge_sd(EI, t, e, s, d);
  float alpha = exBuf[(size_t)t * ET_ + e] / denom[(size_t)t * N_ + d];
  atomicAdd(&outBuf[((size_t)t * N_ + d) * H_ + j],
            alpha * HF[((size_t)t * N_ + s) * H_ + j]);
}

// += bias (per feature), optional relu
__global__ void add_bias_act(float* __restrict__ buf, const float* __restrict__ b,
                             int relu, size_t n) {
  size_t i = (size_t)blockIdx.x * blockDim.x + threadIdx.x;
  if (i >= n) return;
  float v = buf[i] + b[i & (H_ - 1)];
  if (relu) v = fmaxf(v, 0.f);
  buf[i] = v;
}

// ---------------------------------------------------------------------------
// C(MxN) += A(MxK) @ B(NxK)^T  via v_wmma_f32_16x16x4_f32.
// One wave computes a 16x64 strip of C (1 M-tile x 4 N-tiles); K split across
// blockIdx.z with atomic accumulation into pre-zeroed C.
//
// Fragment layouts (ISA 7.12.2, 32-bit matrices, wave32):
//   A 16x4 : lane l -> row M = l&15, K pair = 2*(l>>4)   (one b64 load)
//   B  4x16: mirrored with N rows of B (B stored N x K -> B^T columns)
//   C/D v8f: VGPR i -> M = i + 8*(l>>4), N = l&15
//
// Main K loop is branch-free: k0/kchunk are multiples of 4 and out-of-range
// M/N rows are pointer-clamped (their products land only in D rows/cols that
// the guarded store discards). The K%4 remainder runs once, guarded.
// ---------------------------------------------------------------------------
__global__ __launch_bounds__(32)
void gemm_nt_wmma(const float* __restrict__ A, int lda,
                  const float* __restrict__ B, int ldb,
                  float* __restrict__ C, int ldc,
                  int M, int N, int K, int kchunk) {
  int ng    = blockIdx.x;     // group of 4 N-tiles
  int mtile = blockIdx.y;
  int k0 = blockIdx.z * kchunk;
  int k1 = k0 + kchunk; if (k1 > K) k1 = K;
  if (k0 >= k1) return;

  int lane = threadIdx.x;
  int half = lane >> 4;       // selects K pair {0,1} vs {2,3}
  int l16  = lane & 15;
  int kh   = half * 2;

  int arow = mtile * 16 + l16;
  if (arow >= M) arow = M - 1;                       // clamp; store is guarded
  const float* Ap = A + (size_t)arow * lda;

  const float* Bp[4];
  #pragma unroll
  for (int i = 0; i < 4; ++i) {
    int brow = (ng * 4 + i) * 16 + l16;
    if (brow >= N) brow = N - 1;                     // clamp; store is guarded
    Bp[i] = B + (size_t)brow * ldb;
  }

  v8f acc[4];
  #pragma unroll
  for (int i = 0; i < 4; ++i) acc[i] = (v8f){0.f,0.f,0.f,0.f,0.f,0.f,0.f,0.f};

  int klast = k0 + ((k1 - k0) & ~3);                 // branch-free main range

  #pragma unroll 2
  for (int k = k0; k < klast; k += 4) {
    v2f a = *(const v2f*)(Ap + k + kh);              // global_load_b64
    #pragma unroll
    for (int i = 0; i < 4; ++i) {
      v2f bv = *(const v2f*)(Bp[i] + k + kh);        // global_load_b64
      acc[i] = __builtin_amdgcn_wmma_f32_16x16x4_f32(
          false, a, false, bv, (short)0, acc[i], false, false);
    }
  }

  if (klast < k1) {                                  // K % 4 remainder (once)
    int ka = klast + kh;
    v2f a;
    a.x = (ka     < K) ? Ap[ka]     : 0.f;
    a.y = (ka + 1 < K) ? Ap[ka + 1] : 0.f;
    #pragma unroll
    for (int i = 0; i < 4; ++i) {
      v2f bv;
      bv.x = (ka     < K) ? Bp[i][ka]     : 0.f;
      bv.y = (ka + 1 < K) ? Bp[i][ka + 1] : 0.f;
      acc[i] = __builtin_amdgcn_wmma_f32_16x16x4_f32(
          false, a, false, bv, (short)0, acc[i], false, false);
    }
  }

  #pragma unroll
  for (int i = 0; i < 4; ++i) {
    int ccol = (ng * 4 + i) * 16 + l16;
    if (ccol < N) {
      #pragma unroll
      for (int r = 0; r < 8; ++r) {
        int crow = mtile * 16 + half * 8 + r;
        if (crow < M) atomicAdd(&C[(size_t)crow * ldc + ccol], acc[i][r]);
      }
    }
  }
}

// ---------------------------------------------------------------------------
// LSTM recurrence, one persistent workgroup. Gates precomputed in G (t x 1008).
// gate order (jnp.split): i, f, g, o
// ---------------------------------------------------------------------------
__global__ __launch_bounds__(1024)
void lstm_seq(const float* __restrict__ G, int ldg,
              const float* __restrict__ Whh,
              const float* __restrict__ bih, const float* __restrict__ bhh,
              const float* __restrict__ h0, const float* __restrict__ c0,
              float* __restrict__ ys) {
  __shared__ float h_sh[LH_];
  __shared__ float g_sh[G4_];
  int tid = threadIdx.x;
  float c = 0.f;
  if (tid < LH_) { h_sh[tid] = h0[tid]; c = c0[tid]; }
  __syncthreads();
  for (int t = 0; t < T_; ++t) {
    if (tid < G4_) {
      float acc = G[(size_t)t * ldg + tid] + bih[tid] + bhh[tid];
      const float* wr = Whh + (size_t)tid * LH_;
      for (int k = 0; k < LH_; ++k) acc += wr[k] * h_sh[k];
      g_sh[tid] = acc;
    }
    __syncthreads();
    if (tid < LH_) {
      float ig = g_sh[tid];
      float fg = g_sh[LH_ + tid];
      float gg = g_sh[2 * LH_ + tid];
      float og = g_sh[3 * LH_ + tid];
      float si = 1.f / (1.f + expf(-ig));
      float sf = 1.f / (1.f + expf(-fg));
      float so = 1.f / (1.f + expf(-og));
      c = sf * c + si * tanhf(gg);
      float h = so * tanhf(c);
      ys[(size_t)t * LH_ + tid] = h;
      h_sh[tid] = h;
    }
    __syncthreads();
  }
}

// ---------------------------------------------------------------------------
// 4 FC heads from last hidden state
// ---------------------------------------------------------------------------
__global__ void fc_heads(const float* __restrict__ last,
                         const float* __restrict__ W0, const float* __restrict__ b0,
                         const float* __restrict__ W1, const float* __restrict__ b1,
                         const float* __restrict__ W2, const float* __restrict__ b2,
                         const float* __restrict__ W3, const float* __restrict__ b3,
                         float* __restrict__ out) {
  int o = blockIdx.x * blockDim.x + threadIdx.x;
  int f = blockIdx.y;
  if (o >= OUT_) return;
  const float* W = (f == 0) ? W0 : (f == 1) ? W1 : (f == 2) ? W2 : W3;
  const float* b = (f == 0) ? b0 : (f == 1) ? b1 : (f == 2) ? b2 : b3;
  float acc = b[o];
  const float* wr = W + (size_t)o * LH_;
  for (int k = 0; k < LH_; ++k) acc += wr[k] * last[k];
  out[(size_t)f * OUT_ + o] = acc;
}

// ---------------------------------------------------------------------------
// host
// ---------------------------------------------------------------------------
static inline int cdiv(size_t a, int b) { return (int)((a + b - 1) / b); }

extern "C" void kernel_launch(void* const* d_in, const int* in_sizes, int n_in,
                              void* d_out, int out_size, void* d_ws, size_t ws_size,
                              hipStream_t stream) {
  const float* nodes = (const float*)d_in[0];
  const int*   EI    = (const int*)d_in[1];
  // d_in[2] = edges_attr (unused by reference)
  const float* W1  = (const float*)d_in[3];
  const float* as1 = (const float*)d_in[4];
  const float* ad1 = (const float*)d_in[5];
  const float* b1  = (const float*)d_in[6];
  const float* W2  = (const float*)d_in[7];
  const float* as2 = (const float*)d_in[8];
  const float* ad2 = (const float*)d_in[9];
  const float* b2  = (const float*)d_in[10];
  const float* Wih0 = (const float*)d_in[11];
  const float* Whh0 = (const float*)d_in[12];
  const float* bih0 = (const float*)d_in[13];
  const float* bhh0 = (const float*)d_in[14];
  const float* Wih1 = (const float*)d_in[15];
  const float* Whh1 = (const float*)d_in[16];
  const float* bih1 = (const float*)d_in[17];
  const float* bhh1 = (const float*)d_in[18];
  const float* h0   = (const float*)d_in[19];   // (2,1,250)
  const float* c0   = (const float*)d_in[20];
  const float* fcW1 = (const float*)d_in[21];
  const float* fcb1 = (const float*)d_in[22];
  const float* fcW2 = (const float*)d_in[23];
  const float* fcb2 = (const float*)d_in[24];
  const float* fcW3 = (const float*)d_in[25];
  const float* fcb3 = (const float*)d_in[26];
  const float* fcW4 = (const float*)d_in[27];
  const float* fcb4 = (const float*)d_in[28];
  float* out = (float*)d_out;

  // ---- workspace layout (floats) ----
  const size_t SZ_HF = (size_t)T_ * N_ * H_;      // 3,502,080
  const size_t SZ_TN = (size_t)T_ * N_;           //   109,440
  const size_t SZ_TE = (size_t)T_ * ET_;          // 2,298,240
  const size_t SZ_G  = (size_t)T_ * LDG_;         //    96,768
  const size_t SZ_YS = (size_t)T_ * LH_;          //    24,000

  float* ws = (float*)d_ws;
  float*    HF    = ws;                         // node features (per layer, reused)
  float*    OUT1  = HF    + SZ_HF;              // layer-1 aggregated output
  float*    Xseq  = OUT1  + SZ_HF;              // layer-2 output == LSTM input (96 x 36480)
  float*    ssrc  = Xseq  + SZ_HF;
  float*    sdst  = ssrc  + SZ_TN;
  unsigned* mEnc  = (unsigned*)(sdst + SZ_TN);
  float*    denom = (float*)(mEnc + SZ_TN);
  float*    eBuf  = denom + SZ_TN;
  float*    exBuf = eBuf  + SZ_TE;
  float*    G     = exBuf + SZ_TE;
  float*    ys0   = G     + SZ_G;
  float*    ys1   = ys0   + SZ_YS;
  // total ~15.7M floats (~63 MB) of d_ws

  dim3 bNode(32, 8), gNode(cdiv(N_, 8), T_);
  dim3 bEdge(256),   gEdge(cdiv(ET_, 256), T_);
  dim3 bAgg(32, 8),  gAgg(cdiv(ET_, 8), T_);

  // ============== GAT layer 1 ==============
  fill_f32<<<cdiv(SZ_HF, 256), 256, 0, stream>>>(OUT1, 0.f, SZ_HF);
  fill_u32<<<cdiv(SZ_TN, 256), 256, 0, stream>>>(mEnc, ENC_NEG_INF, SZ_TN);
  fill_f32<<<cdiv(SZ_TN, 256), 256, 0, stream>>>(denom, 0.f, SZ_TN);
  gat_proj<<<gNode, bNode, 0, stream>>>(nodes, F_, W1, as1, ad1, HF, ssrc, sdst);
  edge_score_max<<<gEdge, bEdge, 0, stream>>>(EI, ssrc, sdst, eBuf, mEnc);
  edge_exp<<<gEdge, bEdge, 0, stream>>>(EI, eBuf, mEnc, denom, exBuf);
  edge_agg<<<gAgg, bAgg, 0, stream>>>(EI, exBuf, denom, HF, OUT1);
  add_bias_act<<<cdiv(SZ_HF, 256), 256, 0, stream>>>(OUT1, b1, 1, SZ_HF);

  // ============== GAT layer 2 (aggregates straight into Xseq) ==============
  fill_f32<<<cdiv(SZ_HF, 256), 256, 0, stream>>>(Xseq, 0.f, SZ_HF);
  fill_u32<<<cdiv(SZ_TN, 256), 256, 0, stream>>>(mEnc, ENC_NEG_INF, SZ_TN);
  fill_f32<<<cdiv(SZ_TN, 256), 256, 0, stream>>>(denom, 0.f, SZ_TN);
  gat_proj<<<gNode, bNode, 0, stream>>>(OUT1, H_, W2, as2, ad2, HF, ssrc, sdst);
  edge_score_max<<<gEdge, bEdge, 0, stream>>>(EI, ssrc, sdst, eBuf, mEnc);
  edge_exp<<<gEdge, bEdge, 0, stream>>>(EI, eBuf, mEnc, denom, exBuf);
  edge_agg<<<gAgg, bAgg, 0, stream>>>(EI, exBuf, denom, HF, Xseq);
  add_bias_act<<<cdiv(SZ_HF, 256), 256, 0, stream>>>(Xseq, b2, 0, SZ_HF);

  // ============== LSTM layer 0: G = Xseq @ Wih0^T (the big WMMA GEMM) =======
  {
    const int K = N_ * H_;                 // 36480
    const int KSPLIT = 24;                 // 36480/24 = 1520 (mult of 4)
    fill_f32<<<cdiv(SZ_G, 256), 256, 0, stream>>>(G, 0.f, SZ_G);
    dim3 grid(16, T_ / 16, KSPLIT);        // 16 n-groups x 6 m-tiles x 24 ksplit
    gemm_nt_wmma<<<grid, 32, 0, stream>>>(Xseq, K, Wih0, K, G, LDG_,
                                          T_, G4_, K, K / KSPLIT);
    lstm_seq<<<1, 1024, 0, stream>>>(G, LDG_, Whh0, bih0, bhh0,
                                     h0 + 0, c0 + 0, ys0);
  }

  // ============== LSTM layer 1: G = ys0 @ Wih1^T ==============
  {
    const int K = LH_;                     // 250 (remainder handled in epilogue)
    fill_f32<<<cdiv(SZ_G, 256), 256, 0, stream>>>(G, 0.f, SZ_G);
    dim3 grid(16, T_ / 16, 2);             // kchunk 128: z=1 takes 128..249
    gemm_nt_wmma<<<grid, 32, 0, stream>>>(ys0, K, Wih1, K, G, LDG_,
                                          T_, G4_, K, 128);
    lstm_seq<<<1, 1024, 0, stream>>>(G, LDG_, Whh1, bih1, bhh1,
                                     h0 + LH_, c0 + LH_, ys1);
  }

  // ============== FC heads from ys1[last] ==============
  fc_heads<<<dim3(cdiv(OUT_, 256), 4), 256, 0, stream>>>(
      ys1 + (size_t)(T_ - 1) * LH_,
      fcW1, fcb1, fcW2, fcb2, fcW3, fcb3, fcW4, fcb4, out);

  (void)in_sizes; (void)n_in; (void)out_size; (void)ws_size;
}